// mParametricLIF_523986010139
// MI455X (gfx1250) — compile-verified
//
#include <hip/hip_runtime.h>
#include <hip/hip_bf16.h>

// Momentum parametric LIF forward: x [N,T,D] -> spikes [N,T,D], fp32.
// N=128, T=64, D=4096. Bandwidth-bound (268 MB @ 23.3 TB/s ~ 11.5 us floor).
// gfx1250 path: per-wave 12-deep async global->LDS prefetch ring
// (global_load_async_to_lds_b128 / ASYNCcnt) feeding the sequential T-scan,
// with non-temporal b128 stores for the write-once output stream.

#define N_    128
#define T_    64
#define D_    4096
#define VEC   4
#define BLOCK 256
#define WAVES (BLOCK / 32)
#define NBUF  12                   // prefetch ring depth per wave
#define WAVE_BYTES (32 * 16)       // 512 B staged per wave per timestep

typedef float v4f __attribute__((ext_vector_type(4)));

__global__ __launch_bounds__(BLOCK) void mplif_async_kernel(
    const float* __restrict__ x,
    const float* __restrict__ tau_param,
    const float* __restrict__ lamb,
    const float* __restrict__ th,
    float* __restrict__ out)
{
    // Per-wave private ring: WAVES * NBUF * 512 B = 48 KB (of 320 KB/WGP)
    __shared__ float4 smem[WAVES * NBUF * 32];

    const int tid  = threadIdx.x;
    const int lane = tid & 31;
    const int wave = tid >> 5;

    // 4 blocks per n; each block covers BLOCK*VEC = 1024 consecutive d.
    const int n     = blockIdx.x >> 2;
    const int dbase = (blockIdx.x & 3) * (BLOCK * VEC);
    const int d0    = dbase + tid * VEC;

    const float k   = 1.0f / (1.0f + __expf(-tau_param[0]));
    const float lam = lamb[0];
    const float thv = th[0];

    const float* gsrc = x   + (size_t)n * T_ * D_ + d0;
    float*       gdst = out + (size_t)n * T_ * D_ + d0;

    float4* wbuf = &smem[wave * NBUF * 32];
    // 32-bit LDS byte address of this lane's 16B slot in ring buffer 0
    // (flat shared address truncated to low 32 bits == LDS offset).
    const unsigned lds_lane0 = (unsigned)(size_t)(&wbuf[lane]);

    // ---- prologue: fill the ring (buffer b <- timestep b) ----------------
#pragma unroll
    for (int b = 0; b < NBUF; ++b) {
        unsigned lds_addr = lds_lane0 + b * WAVE_BYTES;
        unsigned long long ga =
            (unsigned long long)(size_t)(gsrc + (size_t)b * D_);
        asm volatile("global_load_async_to_lds_b128 %0, %1, off"
                     :: "v"(lds_addr), "v"(ga) : "memory");
    }

    float4 v = make_float4(0.f, 0.f, 0.f, 0.f);
    float4 m = make_float4(0.f, 0.f, 0.f, 0.f);

#define LIF_STEP(c)                                                  \
    do {                                                             \
        m.c = lam * m.c + k * (xi.c - v.c);                          \
        v.c = v.c + m.c;                                             \
        sp.c = (v.c >= thv) ? 1.0f : 0.0f;                           \
        v.c  = (v.c >= thv) ? 0.0f : v.c;                            \
    } while (0)

    // ---- steady state: consume buffer for t, refill it for t+NBUF --------
    int b = 0;
    for (int t = 0; t < T_ - NBUF; ++t) {
        // In-order async completion: <=11 outstanding => timestep t landed.
        asm volatile("s_wait_asynccnt 11" ::: "memory");

        float4 xi = wbuf[b * 32 + lane];          // ds_load_b128

        float4 sp;
        LIF_STEP(x); LIF_STEP(y); LIF_STEP(z); LIF_STEP(w);

        // LDS read must complete before the async engine overwrites buffer b
        {
            unsigned lds_addr = lds_lane0 + b * WAVE_BYTES;
            unsigned long long ga =
                (unsigned long long)(size_t)(gsrc + (size_t)(t + NBUF) * D_);
            asm volatile("s_wait_dscnt 0\n\t"
                         "global_load_async_to_lds_b128 %0, %1, off"
                         :: "v"(lds_addr), "v"(ga) : "memory");
        }

        // write-once output stream: non-temporal b128 store
        __builtin_nontemporal_store(*(const v4f*)&sp,
                                    (v4f*)(gdst + (size_t)t * D_));

        b = (b + 1 == NBUF) ? 0 : b + 1;          // rotate ring index
    }

    // ---- epilogue: drain the ring ----------------------------------------
    asm volatile("s_wait_asynccnt 0" ::: "memory");
    for (int t = T_ - NBUF; t < T_; ++t) {
        float4 xi = wbuf[b * 32 + lane];

        float4 sp;
        LIF_STEP(x); LIF_STEP(y); LIF_STEP(z); LIF_STEP(w);

        __builtin_nontemporal_store(*(const v4f*)&sp,
                                    (v4f*)(gdst + (size_t)t * D_));

        b = (b + 1 == NBUF) ? 0 : b + 1;
    }
#undef LIF_STEP
}

extern "C" void kernel_launch(void* const* d_in, const int* in_sizes, int n_in,
                              void* d_out, int out_size, void* d_ws, size_t ws_size,
                              hipStream_t stream) {
    (void)in_sizes; (void)n_in; (void)out_size; (void)d_ws; (void)ws_size;

    const float* x   = (const float*)d_in[0];
    const float* tau = (const float*)d_in[1];
    const float* lam = (const float*)d_in[2];
    const float* th  = (const float*)d_in[3];
    float* out       = (float*)d_out;

    // D_/(BLOCK*VEC) = 4 blocks per n, N_ = 128 -> 512 blocks
    dim3 grid(N_ * (D_ / (BLOCK * VEC)));
    dim3 block(BLOCK);
    mplif_async_kernel<<<grid, block, 0, stream>>>(x, tau, lam, th, out);
}